// RNN_25512105738297
// MI455X (gfx1250) — compile-verified
//
#include <hip/hip_runtime.h>
#include <math.h>

// ---------------------------------------------------------------------------
// Single-step Elman RNN on MI455X (gfx1250, wave32).
//   hn     = tanh(x @ W_ih^T + b_ih + h @ W_hh^T + b_hh)   [256,4096]
//   logits = hn @ W_lin^T + b_lin                           [256,4096]
//   probs  = softmax(logits, axis=-1)
// fp32 GEMMs via native V_WMMA_F32_16X16X4_F32. The 192MB fp32 weight stream
// dominates (>= 8.3us at 23.3 TB/s), so global->LDS staging is done with the
// Tensor Data Mover (tensor_load_to_lds, TENSORcnt) double-buffered against
// the WMMA compute. TDM pad_interval/pad_amount reproduce the 36-float padded
// LDS row stride, keeping fragment ds_load_b64 reads bank-conflict free.
// ---------------------------------------------------------------------------

typedef __attribute__((ext_vector_type(2))) float v2f;
typedef __attribute__((ext_vector_type(4))) float v4f;
typedef __attribute__((ext_vector_type(8))) float v8f;
typedef __attribute__((ext_vector_type(4))) unsigned int u32x4;
typedef __attribute__((ext_vector_type(8))) int i32x8;
typedef __attribute__((ext_vector_type(4))) int i32x4;

#if __has_builtin(__builtin_amdgcn_tensor_load_to_lds) && \
    __has_builtin(__builtin_amdgcn_s_wait_tensorcnt)
#define USE_TDM 1
#else
#define USE_TDM 0
#endif

#define B_DIM 256
#define H_DIM 4096
#define K_DIM 4096

#define MT 64                    // M tile per workgroup
#define NT 128                   // N tile per workgroup
#define KC 32                    // K chunk staged in LDS (8 wmma k-steps)
#define LDSS (KC + 4)            // 36: rows 16B-aligned, conflict-free frags
#define NTHREADS 256             // 8 waves

#if USE_TDM
// Low 32 bits of a generic LDS pointer == byte offset within workgroup LDS.
__device__ __forceinline__ unsigned lds_off(const void* p) {
    return (unsigned)(uintptr_t)p;
}

// Issue one TDM DMA: `rows` x KC fp32 tile from a row-major [tdim1, K] matrix
// into LDS with a 4-DWORD pad after every 32 DWORDs (row stride = LDSS = 36).
__device__ __forceinline__ void tdm_issue(const float* gsrc, unsigned ldsOff,
                                          int rows, int K, int tdim1) {
    const unsigned long long ga = (unsigned long long)(uintptr_t)gsrc;
    u32x4 g0;
    g0[0] = 1u;                                               // count=1 (valid D#)
    g0[1] = ldsOff;                                           // lds_addr (bytes)
    g0[2] = (unsigned)(ga & 0xFFFFFFFFu);                     // global_addr[31:0]
    g0[3] = (unsigned)((ga >> 32) & 0x01FFFFFFu) | (2u << 30); // addr[56:32], type=2

    i32x8 g1;
    g1[0] = (int)((2u << 16)      // data_size = 4 bytes
                | (1u << 20)      // pad_enable
                | (4u << 22)      // pad_interval: every 32 DWORDs
                | (3u << 25));    // pad_amount: 4 DWORDs  -> stride 36 floats
    g1[1] = (int)(((unsigned)K & 0xFFFFu) << 16);             // tensor_dim0 lo16
    g1[2] = (int)((((unsigned)K >> 16) & 0xFFFFu)             // tensor_dim0 hi16
                | (((unsigned)tdim1 & 0xFFFFu) << 16));       // tensor_dim1 lo16
    g1[3] = (int)((((unsigned)tdim1 >> 16) & 0xFFFFu)         // tensor_dim1 hi16
                | ((unsigned)KC << 16));                      // tile_dim0 = 32
    g1[4] = rows & 0xFFFF;                                    // tile_dim1; tile_dim2=0
    g1[5] = K;                                                // tensor_dim0_stride lo32
    g1[6] = 0;
    g1[7] = 0;

    const i32x4 z4 = {0, 0, 0, 0};                            // 2D tensor: no grp2/3
    const i32x8 z8 = {0, 0, 0, 0, 0, 0, 0, 0};
    __builtin_amdgcn_tensor_load_to_lds(g0, g1, z4, z4, z8, 0);
}
#else
// Fallback: synchronous coalesced float4 global->LDS tile load.
__device__ __forceinline__ void load_tile(const float* __restrict__ g,
                                          float* __restrict__ lds,
                                          int rows, int row0, int k0,
                                          int ldg, int tid) {
    const int total = rows * (KC / 4);
    for (int f = tid; f < total; f += NTHREADS) {
        const int r  = f >> 3;
        const int c4 = f & 7;
        v4f v = *(const v4f*)(g + (size_t)(row0 + r) * ldg + k0 + c4 * 4);
        *(v4f*)(lds + r * LDSS + c4 * 4) = v;
    }
}
#endif

// Accumulate acc[2][2] (wave's 32x32 patch of the MTxNT tile) over K.
// A is [tdimA, K] row-major, W is [tdimW, K] row-major: computes A @ W^T.
__device__ __forceinline__ void gemm_accum(const float* __restrict__ Ag,
                                           const float* __restrict__ Wg,
                                           float* As0, float* Ws0,
                                           float* As1, float* Ws1,
                                           int m0, int n0, int K,
                                           int tdimA, int tdimW,
                                           v8f acc[2][2],
                                           int tid, int lane,
                                           int waveM, int waveN) {
    const int half = lane >> 4;                 // 0: K{0,1}, 1: K{2,3}
    const int l16  = lane & 15;
    const int nCh  = K / KC;

#if USE_TDM
    if (tid < 32) {                             // wave 0 drives the TDM
        tdm_issue(Ag + (size_t)m0 * K, lds_off(As0), MT, K, tdimA);
        tdm_issue(Wg + (size_t)n0 * K, lds_off(Ws0), NT, K, tdimW);
    }
#endif

    for (int i = 0; i < nCh; ++i) {
        const int k0 = i * KC;
        float* Asb = (i & 1) ? As1 : As0;
        float* Wsb = (i & 1) ? Ws1 : Ws0;
#if USE_TDM
        if (tid < 32) {
            if (i + 1 < nCh) {                  // prefetch next chunk via TDM
                float* Asn = (i & 1) ? As0 : As1;
                float* Wsn = (i & 1) ? Ws0 : Ws1;
                tdm_issue(Ag + (size_t)m0 * K + k0 + KC, lds_off(Asn), MT, K, tdimA);
                tdm_issue(Wg + (size_t)n0 * K + k0 + KC, lds_off(Wsn), NT, K, tdimW);
                __builtin_amdgcn_s_wait_tensorcnt(2);  // current chunk landed
            } else {
                __builtin_amdgcn_s_wait_tensorcnt(0);  // drain pipeline
            }
        }
#else
        load_tile(Ag, Asb, MT, m0, k0, K, tid);
        load_tile(Wg, Wsb, NT, n0, k0, K, tid);
#endif
        __syncthreads();
#pragma unroll
        for (int kk = 0; kk < KC / 4; ++kk) {
            v2f a[2], b[2];
#pragma unroll
            for (int r = 0; r < 2; ++r)
                a[r] = *(const v2f*)&Asb[(waveM * 32 + r * 16 + l16) * LDSS + kk * 4 + half * 2];
#pragma unroll
            for (int c = 0; c < 2; ++c)
                b[c] = *(const v2f*)&Wsb[(waveN * 32 + c * 16 + l16) * LDSS + kk * 4 + half * 2];
#pragma unroll
            for (int r = 0; r < 2; ++r)
#pragma unroll
                for (int c = 0; c < 2; ++c)
                    acc[r][c] = __builtin_amdgcn_wmma_f32_16x16x4_f32(
                        false, a[r], false, b[c], (short)0, acc[r][c], false, false);
        }
        __syncthreads();
    }
}

// Kernel 1: hn = tanh(x@W_ih^T + b_ih + h@W_hh^T + b_hh)
__global__ __launch_bounds__(NTHREADS)
void rnn_hn_kernel(const float* __restrict__ x, const float* __restrict__ h,
                   const float* __restrict__ w_ih, const float* __restrict__ b_ih,
                   const float* __restrict__ w_hh, const float* __restrict__ b_hh,
                   float* __restrict__ hn) {
    __shared__ float As0[MT * LDSS], Ws0[NT * LDSS];
    __shared__ float As1[MT * LDSS], Ws1[NT * LDSS];
    const int tid = threadIdx.x, lane = tid & 31, wave = tid >> 5;
    const int waveM = wave >> 2, waveN = wave & 3;
    const int n0 = blockIdx.x * NT, m0 = blockIdx.y * MT;

    v8f acc[2][2] = {};
    gemm_accum(x, w_ih, As0, Ws0, As1, Ws1, m0, n0, K_DIM, B_DIM, H_DIM,
               acc, tid, lane, waveM, waveN);
    gemm_accum(h, w_hh, As0, Ws0, As1, Ws1, m0, n0, K_DIM, B_DIM, H_DIM,
               acc, tid, lane, waveM, waveN);

    const int half = lane >> 4, l16 = lane & 15;
#pragma unroll
    for (int r = 0; r < 2; ++r)
#pragma unroll
        for (int c = 0; c < 2; ++c) {
            const int gn   = n0 + waveN * 32 + c * 16 + l16;
            const float bn = b_ih[gn] + b_hh[gn];
#pragma unroll
            for (int e = 0; e < 8; ++e) {
                const int gm = m0 + waveM * 32 + r * 16 + half * 8 + e;
                hn[(size_t)gm * H_DIM + gn] = tanhf(acc[r][c][e] + bn);
            }
        }
}

// Kernel 2: logits = hn@W_lin^T + b_lin
__global__ __launch_bounds__(NTHREADS)
void rnn_logits_kernel(const float* __restrict__ hn,
                       const float* __restrict__ w_lin,
                       const float* __restrict__ b_lin,
                       float* __restrict__ logits) {
    __shared__ float As0[MT * LDSS], Ws0[NT * LDSS];
    __shared__ float As1[MT * LDSS], Ws1[NT * LDSS];
    const int tid = threadIdx.x, lane = tid & 31, wave = tid >> 5;
    const int waveM = wave >> 2, waveN = wave & 3;
    const int n0 = blockIdx.x * NT, m0 = blockIdx.y * MT;

    v8f acc[2][2] = {};
    gemm_accum(hn, w_lin, As0, Ws0, As1, Ws1, m0, n0, K_DIM, B_DIM, H_DIM,
               acc, tid, lane, waveM, waveN);

    const int half = lane >> 4, l16 = lane & 15;
#pragma unroll
    for (int r = 0; r < 2; ++r)
#pragma unroll
        for (int c = 0; c < 2; ++c) {
            const int gn   = n0 + waveN * 32 + c * 16 + l16;
            const float bn = b_lin[gn];
#pragma unroll
            for (int e = 0; e < 8; ++e) {
                const int gm = m0 + waveM * 32 + r * 16 + half * 8 + e;
                logits[(size_t)gm * H_DIM + gn] = acc[r][c][e] + bn;
            }
        }
}

// Kernel 3: row softmax over 4096 logits, one block per row.
__global__ __launch_bounds__(NTHREADS)
void softmax_kernel(const float* __restrict__ logits, float* __restrict__ probs) {
    __shared__ float red[NTHREADS];
    const int row = blockIdx.x, tid = threadIdx.x;
    const float* lrow = logits + (size_t)row * H_DIM;

    float m = -3.402823466e38f;
    for (int i = tid; i < H_DIM; i += NTHREADS) m = fmaxf(m, lrow[i]);
    red[tid] = m;
    __syncthreads();
    for (int s = NTHREADS / 2; s > 0; s >>= 1) {
        if (tid < s) red[tid] = fmaxf(red[tid], red[tid + s]);
        __syncthreads();
    }
    m = red[0];
    __syncthreads();

    float sum = 0.0f;
    for (int i = tid; i < H_DIM; i += NTHREADS) sum += __expf(lrow[i] - m);
    red[tid] = sum;
    __syncthreads();
    for (int s = NTHREADS / 2; s > 0; s >>= 1) {
        if (tid < s) red[tid] += red[tid + s];
        __syncthreads();
    }
    const float inv = 1.0f / red[0];

    for (int i = tid; i < H_DIM; i += NTHREADS)
        probs[(size_t)row * H_DIM + i] = __expf(lrow[i] - m) * inv;
}

extern "C" void kernel_launch(void* const* d_in, const int* in_sizes, int n_in,
                              void* d_out, int out_size, void* d_ws, size_t ws_size,
                              hipStream_t stream) {
    const float* x     = (const float*)d_in[0];   // [256,4096]
    const float* h0    = (const float*)d_in[1];   // [1,256,4096]
    const float* w_ih  = (const float*)d_in[2];   // [4096,4096]
    const float* b_ih  = (const float*)d_in[3];   // [4096]
    const float* w_hh  = (const float*)d_in[4];   // [4096,4096]
    const float* b_hh  = (const float*)d_in[5];   // [4096]
    const float* w_lin = (const float*)d_in[6];   // [4096,4096]
    const float* b_lin = (const float*)d_in[7];   // [4096]

    float* probs  = (float*)d_out;                         // [1,256,4096]
    float* hn     = probs + (size_t)B_DIM * H_DIM;         // [1,256,4096]
    float* logits = (float*)d_ws;                          // [256,4096] scratch

    dim3 grid(H_DIM / NT, B_DIM / MT);                     // 32 x 4 workgroups
    rnn_hn_kernel<<<grid, NTHREADS, 0, stream>>>(x, h0, w_ih, b_ih, w_hh, b_hh, hn);
    rnn_logits_kernel<<<grid, NTHREADS, 0, stream>>>(hn, w_lin, b_lin, logits);
    softmax_kernel<<<B_DIM, NTHREADS, 0, stream>>>(logits, probs);
}